// EfficientLSH_58755152609836
// MI455X (gfx1250) — compile-verified
//
#include <hip/hip_runtime.h>
#include <math.h>

// CDNA5 / gfx1250, wave32.
typedef __attribute__((ext_vector_type(2))) float v2f;
typedef __attribute__((ext_vector_type(8))) float v8f;

#define BATCH     4
#define SEQ       4096
#define EMBED_DIM 128
#define KNN       32
#define CHUNK     512           // columns of the distance strip kept in LDS
#define TILES_PER_WAVE 8        // 8 tiles * 16 cols * 4 waves = 512 cols/chunk

// ---------------------------------------------------------------------------
// Kernel 1: squared norms sq[b*S + s] = |e_{b,s}|^2
// ---------------------------------------------------------------------------
__global__ __launch_bounds__(128) void lsh_sqnorm_kernel(const float* __restrict__ emb,
                                                         float* __restrict__ sq) {
    int row = blockIdx.x * blockDim.x + threadIdx.x;
    if (row >= BATCH * SEQ) return;
    const float4* p = (const float4*)(emb + (size_t)row * EMBED_DIM);
    float s = 0.f;
#pragma unroll
    for (int i = 0; i < EMBED_DIM / 4; ++i) {
        float4 v = p[i];
        s += v.x * v.x + v.y * v.y + v.z * v.z + v.w * v.w;
    }
    sq[row] = s;
}

// ---------------------------------------------------------------------------
// Kernel 2: fused distance (WMMA f32 gram) + per-row top-32.
// One workgroup = 16 rows of one batch. 128 threads = 4 waves.
// ---------------------------------------------------------------------------
__global__ __launch_bounds__(128) void lsh_knn_kernel(const float* __restrict__ emb,
                                                      const float* __restrict__ sq,
                                                      int* __restrict__ nbr,
                                                      float* __restrict__ dst) {
    __shared__ float sdist[16 * CHUNK];   // 32 KB distance strip (reused as merge vals)
    __shared__ int   scidx[16 * 8 * KNN]; // 16 KB merge-stage indices

    const int wg      = blockIdx.x;        // 0 .. BATCH*(SEQ/16)-1
    const int b       = wg >> 8;           // / (SEQ/16)
    const int rowBase = (wg & 255) << 4;   // first row of this 16-row strip
    const int tid     = threadIdx.x;
    const int lane    = tid & 31;
    const int waveId  = tid >> 5;
    const int m       = lane & 15;         // M (A) / N (B,C) position
    const int hi      = lane >> 4;         // half-wave: selects K pair (A/B), M+8 (C)

    const float* embB = emb + (size_t)b * SEQ * EMBED_DIM;
    const float* sqB  = sq + b * SEQ;

    // --- A fragments: 16x4 fp32 tiles over K=128, resident in VGPRs.
    // ISA layout: lanes 0-15 hold M=lane,K={4k,4k+1}; lanes 16-31 hold M,K={4k+2,4k+3}.
    v2f afrag[32];
    {
        const float* arow = embB + (size_t)(rowBase + m) * EMBED_DIM + 2 * hi;
#pragma unroll
        for (int kk = 0; kk < 32; ++kk)
            afrag[kk] = *(const v2f*)(arow + 4 * kk);
    }

    // Squared norms of the 8 C-rows this lane owns (VGPR r -> row r or r+8).
    float sqrow[8];
#pragma unroll
    for (int r = 0; r < 8; ++r)
        sqrow[r] = sqB[rowBase + r + 8 * hi];

    // Per-thread running top-k, sorted ascending, kept in registers.
    float tv[KNN];
    int   ti[KNN];
#pragma unroll
    for (int i = 0; i < KNN; ++i) { tv[i] = INFINITY; ti[i] = 0; }

    const int trow = tid >> 3;  // row (0..15) this thread scans in top-k phase
    const int tsub = tid & 7;   // 64-column segment within the chunk

    for (int c = 0; c < SEQ / CHUNK; ++c) {
        // ---- compute 16 x CHUNK distance strip with V_WMMA_F32_16X16X4_F32
#pragma unroll 1
        for (int t = 0; t < TILES_PER_WAVE; ++t) {
            const int colLocal0 = (waveId * TILES_PER_WAVE + t) * 16;
            const int j0        = c * CHUNK + colLocal0;
            const float* brow   = embB + (size_t)(j0 + m) * EMBED_DIM + 2 * hi;

            // Double-buffered B fragments (groups of 8 K-steps) so the next
            // group's global_load_b64s are in flight while the current group's
            // WMMAs execute; dual accumulators break the XDL RAW chain.
            v2f bf[2][8];
#pragma unroll
            for (int i = 0; i < 8; ++i)
                bf[0][i] = *(const v2f*)(brow + 4 * i);

            v8f acc0 = {};
            v8f acc1 = {};
#pragma unroll
            for (int g = 0; g < 4; ++g) {
                if (g < 3) {
#pragma unroll
                    for (int i = 0; i < 8; ++i)
                        bf[(g + 1) & 1][i] = *(const v2f*)(brow + 4 * ((g + 1) * 8 + i));
                }
#pragma unroll
                for (int i = 0; i < 8; ++i) {
                    v2f a  = afrag[g * 8 + i];
                    v2f bb = bf[g & 1][i];
                    // 8 args: (neg_a, A, neg_b, B, c_mod, C, reuse_a, reuse_b)
                    if (i & 1)
                        acc1 = __builtin_amdgcn_wmma_f32_16x16x4_f32(
                            false, a, false, bb, (short)0, acc1, false, false);
                    else
                        acc0 = __builtin_amdgcn_wmma_f32_16x16x4_f32(
                            false, a, false, bb, (short)0, acc0, false, false);
                }
            }

            const float sqc = sqB[j0 + m];
            const int   gj  = j0 + m;
#pragma unroll
            for (int r = 0; r < 8; ++r) {
                const int row = r + 8 * hi;
                float gram = acc0[r] + acc1[r];
                float d2 = sqrow[r] + sqc - 2.0f * gram;
                d2 = fmaxf(d2, 0.0f);
                const int gi = rowBase + row;
                float d = (gi == gj) ? INFINITY : sqrtf(d2);
                sdist[row * CHUNK + colLocal0 + m] = d;
            }
        }
        __syncthreads();

        // ---- top-k scan of this chunk: thread -> (row trow, 64 cols)
        const float* rowp  = &sdist[trow * CHUNK + tsub * 64];
        const int    jbase = c * CHUNK + tsub * 64;
        for (int x = 0; x < 64; ++x) {
            float d = rowp[x];
            if (d < tv[KNN - 1]) {
                float v = d;
                int   id = jbase + x;
#pragma unroll
                for (int s = 0; s < KNN; ++s) {
                    if (v < tv[s]) {
                        float fv = tv[s]; int fi = ti[s];
                        tv[s] = v; ti[s] = id;
                        v = fv; id = fi;
                    }
                }
            }
        }
        __syncthreads();
    }

    // ---- merge the 8 partial top-32 lists per row (reuse sdist for values)
#pragma unroll
    for (int i = 0; i < KNN; ++i) {
        sdist[trow * (8 * KNN) + tsub * KNN + i] = tv[i];
        scidx[trow * (8 * KNN) + tsub * KNN + i] = ti[i];
    }
    __syncthreads();

    if (tid < 16) {
        float fv[KNN];
        int   fi[KNN];
#pragma unroll
        for (int i = 0; i < KNN; ++i) { fv[i] = INFINITY; fi[i] = 0; }
        for (int x = 0; x < 8 * KNN; ++x) {
            float d = sdist[tid * (8 * KNN) + x];
            if (d < fv[KNN - 1]) {
                float v = d;
                int   id = scidx[tid * (8 * KNN) + x];
#pragma unroll
                for (int s = 0; s < KNN; ++s) {
                    if (v < fv[s]) {
                        float a = fv[s]; int c2 = fi[s];
                        fv[s] = v; fi[s] = id;
                        v = a; id = c2;
                    }
                }
            }
        }
        const size_t orow = ((size_t)b * SEQ + rowBase + tid) * KNN;
#pragma unroll
        for (int i = 0; i < KNN; ++i) {
            nbr[orow + i] = fi[i];   // neighbors: int32 (first output)
            dst[orow + i] = fv[i];   // distances: fp32 (second output)
        }
    }
}

// ---------------------------------------------------------------------------
extern "C" void kernel_launch(void* const* d_in, const int* in_sizes, int n_in,
                              void* d_out, int out_size, void* d_ws, size_t ws_size,
                              hipStream_t stream) {
    const float* emb = (const float*)d_in[0];
    // d_in[1] (projections): dead code in the reference forward — skipped.
    // d_in[2] (k): fixed at 32 by the setup; hardcoded as KNN.
    (void)in_sizes; (void)n_in; (void)out_size; (void)ws_size;

    float* sqnorm = (float*)d_ws;                                // BATCH*SEQ floats
    int*   nbr    = (int*)d_out;                                 // [B,S,32] int32
    float* dst    = (float*)d_out + (size_t)BATCH * SEQ * KNN;   // [B,S,32] f32

    lsh_sqnorm_kernel<<<(BATCH * SEQ + 127) / 128, 128, 0, stream>>>(emb, sqnorm);
    lsh_knn_kernel<<<BATCH * (SEQ / 16), 128, 0, stream>>>(emb, sqnorm, nbr, dst);
}